// GATClassifier_77403900609086
// MI455X (gfx1250) — compile-verified
//
#include <hip/hip_runtime.h>
#include <hip/hip_bf16.h>
#include <math.h>

#define NFEAT 128
#define OUTF  64
#define NEG_SLOPE 0.2f

typedef __attribute__((ext_vector_type(2))) float v2f;
typedef __attribute__((ext_vector_type(8))) float v8f;

// ---------------------------------------------------------------------------
// helpers
// ---------------------------------------------------------------------------
__device__ __forceinline__ float waveReduceSum(float v) {
#pragma unroll
  for (int o = 16; o > 0; o >>= 1) v += __shfl_xor(v, o, 32);
  return v;
}

// atomic max on float via signed/unsigned integer atomics (works for all signs,
// including the -inf init value 0xFF800000)
__device__ __forceinline__ void atomicMaxFloat(float* addr, float v) {
  if (v >= 0.0f) {
    atomicMax((int*)addr, __float_as_int(v));
  } else {
    atomicMin((unsigned int*)addr, __float_as_uint(v));
  }
}

// ---------------------------------------------------------------------------
// 0) init workspace: agg = 0, mx = -inf, den = 0
// ---------------------------------------------------------------------------
__global__ __launch_bounds__(256) void k_init(float* __restrict__ agg,
                                              float* __restrict__ mx,
                                              float* __restrict__ den, int N) {
  int i = blockIdx.x * 256 + threadIdx.x;
  if (i < N * OUTF) agg[i] = 0.0f;
  if (i < N) {
    mx[i]  = __int_as_float(0xff800000);  // -inf
    den[i] = 0.0f;
  }
}

// ---------------------------------------------------------------------------
// 1) fused dual GEMM:  x_l = x @ W_l^T + b_l ; x_r = x @ W_r^T + b_r
//    f32 WMMA 16x16x4. One wave = one 16-row tile of x; it produces all
//    8 output tiles (4 n-tiles x 2 weight matrices), K=128 in 32 steps.
//    A frag (16x4 f32): lane<16 -> rows M=lane, K = k0+{0,1};
//                       lane>=16 -> rows M=lane-16, K = k0+{2,3}.
//    B frag (4x16 f32): VGPR v, lanes<16 -> K=k0+v; lanes>=16 -> K=k0+v+2,
//                       N = lane&15.  B[k][n] = W[n][k]  (for x @ W^T).
//    C/D (16x16 f32): VGPR r -> row M = r + 8*(lane>=16), N = lane&15.
// ---------------------------------------------------------------------------
__global__ __launch_bounds__(256) void k_gemm(const float* __restrict__ x,
                                              const float* __restrict__ Wl,
                                              const float* __restrict__ bl,
                                              const float* __restrict__ Wr,
                                              const float* __restrict__ br,
                                              float* __restrict__ xl,
                                              float* __restrict__ xr,
                                              int nTiles) {
  __shared__ float ldsW[2 * OUTF * NFEAT];  // 64 KB: [W_l ; W_r] row-major

  // cooperative weight stage into LDS (float4 granularity)
  {
    const int nvec = 2 * OUTF * NFEAT / 4;  // 4096 float4
    float4* l4 = (float4*)ldsW;
    for (int i = threadIdx.x; i < nvec; i += 256) {
      l4[i] = (i < OUTF * NFEAT / 4)
                  ? ((const float4*)Wl)[i]
                  : ((const float4*)Wr)[i - OUTF * NFEAT / 4];
    }
  }
  __syncthreads();

  const int wave = threadIdx.x >> 5;
  const int lane = threadIdx.x & 31;
  const int ln   = lane & 15;
  const int hi   = lane >> 4;  // 0: lanes 0-15, 1: lanes 16-31

  const int tile = blockIdx.x * 8 + wave;
  if (tile >= nTiles) return;  // wave-uniform: EXEC stays all-ones for WMMA
  const int m0 = tile * 16;

  v8f acc[8];
  v8f zero = {0.f, 0.f, 0.f, 0.f, 0.f, 0.f, 0.f, 0.f};
#pragma unroll
  for (int t = 0; t < 8; ++t) acc[t] = zero;

  const float* xrow = x + (size_t)(m0 + ln) * NFEAT + 2 * hi;

  for (int k0 = 0; k0 < NFEAT; k0 += 4) {
    v2f a = *(const v2f*)(xrow + k0);  // A frag: K = k0+2*hi, k0+2*hi+1
#pragma unroll
    for (int w = 0; w < 2; ++w) {
#pragma unroll
      for (int nt = 0; nt < 4; ++nt) {
        // B frag: two consecutive K per lane-half of row (nt*16+ln) of W
        const float* wb =
            ldsW + w * OUTF * NFEAT + (nt * 16 + ln) * NFEAT + k0 + 2 * hi;
        v2f b = *(const v2f*)wb;
        acc[w * 4 + nt] = __builtin_amdgcn_wmma_f32_16x16x4_f32(
            /*neg_a=*/false, a, /*neg_b=*/false, b,
            /*c_mod=*/(short)0, acc[w * 4 + nt],
            /*reuse_a=*/false, /*reuse_b=*/false);
      }
    }
  }

#pragma unroll
  for (int w = 0; w < 2; ++w) {
    float* dst        = (w == 0) ? xl : xr;
    const float* bias = (w == 0) ? bl : br;
#pragma unroll
    for (int nt = 0; nt < 4; ++nt) {
      const int col  = nt * 16 + ln;
      const float bv = bias[col];
#pragma unroll
      for (int r = 0; r < 8; ++r) {
        dst[(size_t)(m0 + r + 8 * hi) * OUTF + col] = acc[w * 4 + nt][r] + bv;
      }
    }
  }
}

// scalar tail for rows not covered by full 16-row tiles (N%16, normally 0)
__global__ __launch_bounds__(256) void k_gemm_tail(
    const float* __restrict__ x, const float* __restrict__ Wl,
    const float* __restrict__ bl, const float* __restrict__ Wr,
    const float* __restrict__ br, float* __restrict__ xl,
    float* __restrict__ xr, int rowStart, int N) {
  int idx = blockIdx.x * 256 + threadIdx.x;
  int row = rowStart + idx / OUTF;
  int col = idx % OUTF;
  if (row >= N) return;
  float sl = 0.f, sr = 0.f;
  for (int k = 0; k < NFEAT; ++k) {
    float xv = x[(size_t)row * NFEAT + k];
    sl += xv * Wl[col * NFEAT + k];
    sr += xv * Wr[col * NFEAT + k];
  }
  xl[(size_t)row * OUTF + col] = sl + bl[col];
  xr[(size_t)row * OUTF + col] = sr + br[col];
}

// ---------------------------------------------------------------------------
// 2) edge attention scores + segment max.  Wave-per-edge (wave32), 2 feats/lane.
//    Edges e < E come from edge_index; e >= E are self-loops (src=dst=e-E).
// ---------------------------------------------------------------------------
__global__ __launch_bounds__(256) void k_edge_score(
    const int* __restrict__ ei, const float* __restrict__ xl,
    const float* __restrict__ xr, const float* __restrict__ att,
    float* __restrict__ score, float* __restrict__ mx, int E, int ET) {
  int e = blockIdx.x * 8 + (threadIdx.x >> 5);
  if (e >= ET) return;
  int lane = threadIdx.x & 31;
  int src, dst;
  if (e < E) {
    src = ei[e];
    dst = ei[E + e];
  } else {
    src = dst = e - E;
  }
  float a0 = xl[(size_t)src * OUTF + lane] + xr[(size_t)dst * OUTF + lane];
  float a1 = xl[(size_t)src * OUTF + 32 + lane] + xr[(size_t)dst * OUTF + 32 + lane];
  a0 = (a0 > 0.f) ? a0 : NEG_SLOPE * a0;  // LeakyReLU
  a1 = (a1 > 0.f) ? a1 : NEG_SLOPE * a1;
  float p = att[lane] * a0 + att[lane + 32] * a1;
  p = waveReduceSum(p);
  if (lane == 0) {
    score[e] = p;
    atomicMaxFloat(&mx[dst], p);
  }
}

// ---------------------------------------------------------------------------
// 3) num = exp(score - mx[dst]); den[dst] += num   (thread-per-edge; in-place)
// ---------------------------------------------------------------------------
__global__ __launch_bounds__(256) void k_edge_num(const int* __restrict__ ei,
                                                  float* __restrict__ score,
                                                  const float* __restrict__ mx,
                                                  float* __restrict__ den,
                                                  int E, int ET) {
  int e = blockIdx.x * 256 + threadIdx.x;
  if (e >= ET) return;
  int dst = (e < E) ? ei[E + e] : (e - E);
  float num = __expf(score[e] - mx[dst]);
  score[e]  = num;  // overwrite score with numerator
  atomicAdd(&den[dst], num);
}

// ---------------------------------------------------------------------------
// 4) agg[dst] += alpha * x_l[src]   (wave-per-edge, atomic f32 scatter-add)
// ---------------------------------------------------------------------------
__global__ __launch_bounds__(256) void k_edge_agg(
    const int* __restrict__ ei, const float* __restrict__ num,
    const float* __restrict__ den, const float* __restrict__ xl,
    float* __restrict__ agg, int E, int ET) {
  int e = blockIdx.x * 8 + (threadIdx.x >> 5);
  if (e >= ET) return;
  int lane = threadIdx.x & 31;
  int src, dst;
  if (e < E) {
    src = ei[e];
    dst = ei[E + e];
  } else {
    src = dst = e - E;
  }
  float alpha = num[e] / (den[dst] + 1e-16f);
  atomicAdd(&agg[(size_t)dst * OUTF + lane],
            alpha * xl[(size_t)src * OUTF + lane]);
  atomicAdd(&agg[(size_t)dst * OUTF + 32 + lane],
            alpha * xl[(size_t)src * OUTF + 32 + lane]);
}

// ---------------------------------------------------------------------------
// 5) head: h = relu(agg + bias); out = sigmoid(h . W_lin + b_lin)  (wave/node)
// ---------------------------------------------------------------------------
__global__ __launch_bounds__(256) void k_head(const float* __restrict__ agg,
                                              const float* __restrict__ bias,
                                              const float* __restrict__ wlin,
                                              const float* __restrict__ blin,
                                              float* __restrict__ out, int N) {
  int i = blockIdx.x * 8 + (threadIdx.x >> 5);
  if (i >= N) return;
  int lane = threadIdx.x & 31;
  float h0 = fmaxf(agg[(size_t)i * OUTF + lane] + bias[lane], 0.f);
  float h1 = fmaxf(agg[(size_t)i * OUTF + 32 + lane] + bias[lane + 32], 0.f);
  float p  = h0 * wlin[lane] + h1 * wlin[lane + 32];
  p = waveReduceSum(p);
  if (lane == 0) {
    float z = p + blin[0];
    out[i]  = 1.0f / (1.0f + __expf(-z));
  }
}

// ---------------------------------------------------------------------------
// launcher
// ---------------------------------------------------------------------------
extern "C" void kernel_launch(void* const* d_in, const int* in_sizes, int n_in,
                              void* d_out, int out_size, void* d_ws,
                              size_t ws_size, hipStream_t stream) {
  const float* x    = (const float*)d_in[0];
  const int*   ei   = (const int*)d_in[1];  // [2, E] int32, row-major
  const float* Wl   = (const float*)d_in[2];
  const float* bl   = (const float*)d_in[3];
  const float* Wr   = (const float*)d_in[4];
  const float* br   = (const float*)d_in[5];
  const float* att  = (const float*)d_in[6];
  const float* bias = (const float*)d_in[7];
  const float* Wlin = (const float*)d_in[8];
  const float* blin = (const float*)d_in[9];
  float* out = (float*)d_out;

  const int N  = in_sizes[0] / NFEAT;  // 50000
  const int E  = in_sizes[1] / 2;      // 800000
  const int ET = E + N;                // edges + self loops

  // workspace layout (floats)
  float* ws    = (float*)d_ws;
  size_t off   = 0;
  float* xl    = ws + off; off += (size_t)N * OUTF;
  float* xr    = ws + off; off += (size_t)N * OUTF;
  float* agg   = ws + off; off += (size_t)N * OUTF;
  float* mx    = ws + off; off += (size_t)N;
  float* den   = ws + off; off += (size_t)N;
  float* score = ws + off; off += (size_t)ET;  // reused as numerator
  (void)ws_size; (void)n_in; (void)out_size;

  // 0) init
  {
    int total = N * OUTF;
    k_init<<<(total + 255) / 256, 256, 0, stream>>>(agg, mx, den, N);
  }
  // 1) dual GEMM via f32 WMMA
  {
    int nTiles = N / 16;
    if (nTiles > 0)
      k_gemm<<<(nTiles + 7) / 8, 256, 0, stream>>>(x, Wl, bl, Wr, br, xl, xr,
                                                   nTiles);
    int rem = N - nTiles * 16;
    if (rem > 0)
      k_gemm_tail<<<(rem * OUTF + 255) / 256, 256, 0, stream>>>(
          x, Wl, bl, Wr, br, xl, xr, nTiles * 16, N);
  }
  // 2) scores + segment max
  k_edge_score<<<(ET + 7) / 8, 256, 0, stream>>>(ei, xl, xr, att, score, mx, E,
                                                 ET);
  // 3) exp + segment sum
  k_edge_num<<<(ET + 255) / 256, 256, 0, stream>>>(ei, score, mx, den, E, ET);
  // 4) weighted scatter-add aggregation
  k_edge_agg<<<(ET + 7) / 8, 256, 0, stream>>>(ei, score, den, xl, agg, E, ET);
  // 5) relu + linear + sigmoid head
  k_head<<<(N + 7) / 8, 256, 0, stream>>>(agg, bias, Wlin, blin, out, N);
}